// SPEmbedder3ConvUNIVERSAL_21062519620289
// MI455X (gfx1250) — compile-verified
//
#include <hip/hip_runtime.h>

typedef __attribute__((ext_vector_type(2))) float v2f;
typedef __attribute__((ext_vector_type(8))) float v8f;

#define HID 128
#define SLOPE 0.01f
#define EPSN 1e-5f

__device__ __forceinline__ float leakyf(float v) { return v > 0.f ? v : SLOPE * v; }

// Monotone float -> uint encoding so atomicMax/Min on uint == float max/min.
__device__ __forceinline__ unsigned fenc(float f) {
    unsigned u = __float_as_uint(f);
    return (u & 0x80000000u) ? ~u : (u | 0x80000000u);
}
__device__ __forceinline__ float fdec(unsigned e) {
    unsigned u = (e & 0x80000000u) ? (e & 0x7FFFFFFFu) : ~e;
    return __uint_as_float(u);
}

// ---------------- generic fill ----------------
__global__ void fill_u32_kernel(unsigned* p, unsigned v, int n) {
    int i = blockIdx.x * blockDim.x + threadIdx.x;
    int s = gridDim.x * blockDim.x;
    for (; i < n; i += s) p[i] = v;
}

// ---------------- degrees / counts ----------------
__global__ void degree_kernel(const int* __restrict__ src, const int* __restrict__ dst,
                              float* __restrict__ degOut, float* __restrict__ degIn, int E) {
    int i = blockIdx.x * blockDim.x + threadIdx.x;
    int s = gridDim.x * blockDim.x;
    for (; i < E; i += s) {
        atomicAdd(&degOut[src[i]], 1.f);
        atomicAdd(&degIn[dst[i]], 1.f);
    }
}

__global__ void rsqrt_deg_kernel(float* cs, float* cd, int N) {
    int i = blockIdx.x * blockDim.x + threadIdx.x;
    int s = gridDim.x * blockDim.x;
    for (; i < N; i += s) {
        cs[i] = rsqrtf(fmaxf(cs[i], 1.f));
        cd[i] = rsqrtf(fmaxf(cd[i], 1.f));
    }
}

__global__ void count_kernel(const int* __restrict__ gids, int* __restrict__ cnt, int N) {
    int i = blockIdx.x * blockDim.x + threadIdx.x;
    int s = gridDim.x * blockDim.x;
    for (; i < N; i += s) atomicAdd(&cnt[gids[i]], 1);
}

// ---------------- GEMM via V_WMMA_F32_16X16X4_F32, fused cs scaling ----------------
// hs[n, 0:128] = (x[n, 0:K] @ W[K,128]) * cs[n]
// W is staged into LDS in k-pair-interleaved layout:
//   wlds[(k>>1)*256 + col*2 + (k&1)] = W[k, col]
// so the B fragment (W[kk,col], W[kk+1,col]) for even kk is one aligned 8-byte LDS load.
__global__ __launch_bounds__(256) void gemm_cs_kernel(const float* __restrict__ x,
                                                      const float* __restrict__ W,
                                                      const float* __restrict__ cs,
                                                      float* __restrict__ hs, int N, int K) {
    __shared__ float wlds[128 * HID];
    int tid = threadIdx.x;
    int tot = K * HID;
    for (int i = tid; i < tot; i += 256) {
        int k = i >> 7;        // row of W
        int col = i & 127;     // column of W
        wlds[(k >> 1) * (2 * HID) + col * 2 + (k & 1)] = W[i];
    }
    __syncthreads();

    int wv = tid >> 5;
    int lane = tid & 31;
    int rowTile = blockIdx.x * 8 + wv;
    int numTiles = (N + 15) >> 4;
    if (rowTile >= numTiles) return;  // wave-uniform: WMMA below runs with full EXEC
    int rowBase = rowTile << 4;

    int lo = lane & 15;
    int hi = lane >> 4;
    int ksub = hi * 2;  // lanes 0-15 carry k,k+1 ; lanes 16-31 carry k+2,k+3

    v8f acc[8] = {};

    int arow = rowBase + lo;
    if (arow > N - 1) arow = N - 1;  // clamp (tail safety; N=50000 is exact)
    const float* xr = x + (size_t)arow * K;

    // per-lane base into the interleaved LDS W: row-pair (k+ksub)>>1, column col*2
    const float* wbase = wlds + (size_t)(ksub >> 1) * (2 * HID) + lo * 2;

    for (int k = 0; k < K; k += 4) {
        v2f a = *reinterpret_cast<const v2f*>(xr + k + ksub);  // one global_load_b64
        const float* wk = wbase + (size_t)(k >> 1) * (2 * HID);
        v2f b[8];
#pragma unroll
        for (int ct = 0; ct < 8; ct++)  // 8 back-to-back ds_load_b64, no packing movs
            b[ct] = *reinterpret_cast<const v2f*>(wk + ct * 32);
#pragma unroll
        for (int ct = 0; ct < 8; ct++)
            acc[ct] = __builtin_amdgcn_wmma_f32_16x16x4_f32(
                false, a, false, b[ct], (short)0, acc[ct], false, false);
    }

#pragma unroll
    for (int j = 0; j < 8; j++) {
        int row = rowBase + j + 8 * hi;  // C layout: VGPR j -> rows j (lo half), j+8 (hi half)
        if (row < N) {
            float csv = cs[row];
#pragma unroll
            for (int ct = 0; ct < 8; ct++)
                hs[(size_t)row * HID + ct * 16 + lo] = acc[ct][j] * csv;
        }
    }
}

// ---------------- edge scatter: agg[dst] += hs[src] ----------------
__global__ void scatter_kernel(const float* __restrict__ hs, const int* __restrict__ src,
                               const int* __restrict__ dst, float* __restrict__ agg, int E) {
    int t = blockIdx.x * blockDim.x + threadIdx.x;
    int e = t >> 5;  // one wave per edge: 32 lanes x 4 floats = 128 feats
    if (e >= E) return;
    int c = (t & 31) << 2;
    int sn = src[e], dn = dst[e];
    const float4 v = *reinterpret_cast<const float4*>(hs + (size_t)sn * HID + c);
    float* p = agg + (size_t)dn * HID + c;
    atomicAdd(p + 0, v.x);
    atomicAdd(p + 1, v.y);
    atomicAdd(p + 2, v.z);
    atomicAdd(p + 3, v.w);
}

// ---------------- conv finish (cd scale) + column-sum for GraphNorm mean ----------------
__global__ __launch_bounds__(256) void conv_colsum_kernel(const float* __restrict__ agg,
                                                          const float* __restrict__ cd,
                                                          float* __restrict__ conv,
                                                          float* __restrict__ colsum, int total) {
    __shared__ float ls[HID];
    if (threadIdx.x < HID) ls[threadIdx.x] = 0.f;
    __syncthreads();
    int i0 = blockIdx.x * blockDim.x + threadIdx.x;
    int s = gridDim.x * blockDim.x;  // multiple of 128 -> feature index is loop-invariant
    float part = 0.f;
    for (int j = i0; j < total; j += s) {
        float v = agg[j] * cd[j >> 7];
        conv[j] = v;
        part += v;
    }
    atomicAdd(&ls[threadIdx.x & 127], part);
    __syncthreads();
    if (threadIdx.x < HID) atomicAdd(&colsum[threadIdx.x], ls[threadIdx.x]);
}

// ---------------- center (x - s*mean) + column sum of squares ----------------
__global__ __launch_bounds__(256) void center_kernel(float* __restrict__ conv,
                                                     const float* __restrict__ gscale,
                                                     const float* __restrict__ colsum,
                                                     float* __restrict__ colsum2, int total,
                                                     float invN) {
    __shared__ float ls[HID];
    if (threadIdx.x < HID) ls[threadIdx.x] = 0.f;
    __syncthreads();
    int f = threadIdx.x & 127;
    float sm = gscale[f] * colsum[f] * invN;
    int i0 = blockIdx.x * blockDim.x + threadIdx.x;
    int s = gridDim.x * blockDim.x;
    float part = 0.f;
    for (int j = i0; j < total; j += s) {
        float v = conv[j] - sm;
        conv[j] = v;
        part += v * v;
    }
    atomicAdd(&ls[f], part);
    __syncthreads();
    if (threadIdx.x < HID) atomicAdd(&colsum2[threadIdx.x], ls[threadIdx.x]);
}

// ---------------- normalize+affine+leaky (in place) fused with readout accumulation ----------------
__global__ __launch_bounds__(256) void apply_readout_kernel(
    float* __restrict__ h, const float* __restrict__ w, const float* __restrict__ b,
    const float* __restrict__ colsum2, const int* __restrict__ gids, float* __restrict__ rsum,
    unsigned* __restrict__ rmax, unsigned* __restrict__ rmin, int total, float invN) {
    int f = threadIdx.x & 127;
    float alpha = rsqrtf(colsum2[f] * invN + EPSN) * w[f];
    float beta = b[f];
    int i0 = blockIdx.x * blockDim.x + threadIdx.x;
    int s = gridDim.x * blockDim.x;
    for (int j = i0; j < total; j += s) {
        float v = leakyf(fmaf(h[j], alpha, beta));
        h[j] = v;
        int g = gids[j >> 7];
        int idx = g * HID + f;
        atomicAdd(&rsum[idx], v);
        atomicMax(&rmax[idx], fenc(v));
        atomicMin(&rmin[idx], fenc(v));
    }
}

// ---------------- readout finalize: mean|max|min -> out slice, final leaky ----------------
__global__ void finalize_kernel(const float* __restrict__ rsum, const unsigned* __restrict__ rmax,
                                const unsigned* __restrict__ rmin, const int* __restrict__ cnt,
                                float* __restrict__ out, int G, int layer) {
    int i = blockIdx.x * blockDim.x + threadIdx.x;
    if (i >= G * HID) return;
    int g = i >> 7, f = i & 127;
    int c = cnt[g];
    float mean = c > 0 ? rsum[i] / (float)c : 0.f;
    float mx = c > 0 ? fdec(rmax[i]) : 0.f;
    float mn = c > 0 ? fdec(rmin[i]) : 0.f;
    float* row = out + (size_t)g * (9 * HID) + layer * (3 * HID);
    row[f] = leakyf(mean);
    row[HID + f] = leakyf(mx);
    row[2 * HID + f] = leakyf(mn);
}

extern "C" void kernel_launch(void* const* d_in, const int* in_sizes, int n_in, void* d_out,
                              int out_size, void* d_ws, size_t ws_size, hipStream_t stream) {
    (void)n_in;
    (void)ws_size;
    const float* node_feats = (const float*)d_in[0];
    const float* Wmat[3] = {(const float*)d_in[1], (const float*)d_in[2], (const float*)d_in[3]};
    const float* gw = (const float*)d_in[4];
    const float* gb = (const float*)d_in[5];
    const float* gs = (const float*)d_in[6];
    const int* src = (const int*)d_in[7];
    const int* dst = (const int*)d_in[8];
    const int* gids = (const int*)d_in[9];

    const int N = in_sizes[0] / 64;
    const int E = in_sizes[7];
    const int G = out_size / (9 * HID);
    float* out = (float*)d_out;

    float* ws = (float*)d_ws;
    size_t off = 0;
    float* bufA = ws + off; off += (size_t)N * HID;
    float* bufB = ws + off; off += (size_t)N * HID;  // agg
    float* bufC = ws + off; off += (size_t)N * HID;
    float* cs = ws + off; off += N;
    float* cd = ws + off; off += N;
    int* cnt = (int*)(ws + off); off += G;
    float* colsum = ws + off; off += HID;
    float* colsum2 = ws + off; off += HID;
    float* rsum = ws + off; off += (size_t)G * HID;
    unsigned* rmax = (unsigned*)(ws + off); off += (size_t)G * HID;
    unsigned* rmin = (unsigned*)(ws + off); off += (size_t)G * HID;

    const int total = N * HID;
    const float invN = 1.f / (float)N;
    const int TB = 256;

    // degrees + per-graph counts (cs,cd,cnt are contiguous -> single zero fill)
    fill_u32_kernel<<<(2 * N + G + TB - 1) / TB, TB, 0, stream>>>((unsigned*)cs, 0u, 2 * N + G);
    degree_kernel<<<(E + TB - 1) / TB, TB, 0, stream>>>(src, dst, cs, cd, E);
    rsqrt_deg_kernel<<<(N + TB - 1) / TB, TB, 0, stream>>>(cs, cd, N);
    count_kernel<<<(N + TB - 1) / TB, TB, 0, stream>>>(gids, cnt, N);

    const float* xs[3] = {node_feats, bufA, bufC};
    float* hsb[3] = {bufA, bufC, bufA};  // hs buffer; conv overwrites it in place
    const int Ks[3] = {64, HID, HID};

    const int gemmGrid = (((N + 15) >> 4) + 7) / 8;
    const int gsBlocks = 1024;  // grid-stride elementwise/reduction kernels
    const int statN = 2 * HID + 2 * G * HID;  // colsum, colsum2, rsum, rmax (contiguous)

    for (int layer = 0; layer < 3; layer++) {
        gemm_cs_kernel<<<gemmGrid, TB, 0, stream>>>(xs[layer], Wmat[layer], cs, hsb[layer], N,
                                                    Ks[layer]);
        fill_u32_kernel<<<(total + TB - 1) / TB, TB, 0, stream>>>((unsigned*)bufB, 0u, total);
        fill_u32_kernel<<<(statN + TB - 1) / TB, TB, 0, stream>>>((unsigned*)colsum, 0u, statN);
        fill_u32_kernel<<<(G * HID + TB - 1) / TB, TB, 0, stream>>>(rmin, 0xFFFFFFFFu, G * HID);

        scatter_kernel<<<((size_t)E * 32 + TB - 1) / TB, TB, 0, stream>>>(hsb[layer], src, dst,
                                                                          bufB, E);
        conv_colsum_kernel<<<gsBlocks, TB, 0, stream>>>(bufB, cd, hsb[layer], colsum, total);
        center_kernel<<<gsBlocks, TB, 0, stream>>>(hsb[layer], gs + layer * HID, colsum, colsum2,
                                                   total, invN);
        apply_readout_kernel<<<gsBlocks, TB, 0, stream>>>(hsb[layer], gw + layer * HID,
                                                          gb + layer * HID, colsum2, gids, rsum,
                                                          rmax, rmin, total, invN);
        finalize_kernel<<<(G * HID + TB - 1) / TB, TB, 0, stream>>>(rsum, rmax, rmin, cnt, out, G,
                                                                    layer);
    }
}